// HolonomyComputation_12661563589245
// MI455X (gfx1250) — compile-verified
//
#include <hip/hip_runtime.h>

typedef __attribute__((ext_vector_type(2))) float f32x2;
typedef __attribute__((ext_vector_type(8))) float f32x8;

// Gamma: [4, 512, 512, 8, 8] f32 ; loops: [64, 17] int32 ; out: [4, 64, 8, 8] f32
#define BATCH    4
#define SEQ      512
#define KDIM     8
#define N_LOOPS  64
#define LOOP_LEN 17

__global__ __launch_bounds__(256) void holonomy_wmma_kernel(
    const float* __restrict__ Gamma,
    const int*   __restrict__ loops,
    float*       __restrict__ out)
{
    const int lane = threadIdx.x & 31;
    // one wave per (batch, loop) chain; 8 waves per 256-thread block.
    // wid / b / l are wave-uniform: pin them to SGPRs for scalar addressing.
    const int wid = __builtin_amdgcn_readfirstlane(
        blockIdx.x * 8 + (threadIdx.x >> 5));
    const int b = wid >> 6;   // 0..3
    const int l = wid & 63;   // 0..63

    const int* lp = loops + l * LOOP_LEN;

    // A / B lane mapping for V_WMMA_F32_16X16X4_F32
    const int  mrow = lane & 15;            // row (A) / col (B) index 0..15
    const int  kb   = (lane & 16) ? 2 : 0;  // K sub-block held by this half-wave
    const int  mr   = mrow & 7;             // clamped row (keeps loads in-bounds;
                                            // duplicated rows 8..15 never
                                            // contaminate the valid 8x8 block)
    const bool hi   = (lane >= 16);

    // per-lane constant element offset into one gathered 8x8 tile
    const int voff = mr * KDIM + kb;

    // wave-uniform base offset for this batch (elements)
    const size_t sbase = (size_t)b * SEQ * SEQ * (KDIM * KDIM);

    // acc = 16x16 C/D layout; valid 8x8 block = VGPR r (row), lanes 0..7 (col).
    // Identity in the valid block, zero elsewhere.
    f32x8 acc;
#pragma unroll
    for (int v = 0; v < 8; ++v)
        acc[v] = (lane == v) ? 1.0f : 0.0f;

    int j = __builtin_amdgcn_readfirstlane(lp[0]);
#pragma unroll
    for (int t = 0; t < LOOP_LEN - 1; ++t) {
        const int i = __builtin_amdgcn_readfirstlane(lp[t + 1]);
        const float* gp = Gamma + sbase + (((size_t)i * SEQ + j) << 6);

        // Prefetch next gathered tile (address known one step ahead).
        if (t < LOOP_LEN - 2) {
            const int in = __builtin_amdgcn_readfirstlane(lp[t + 2]);
            __builtin_prefetch(
                Gamma + sbase + (((size_t)in * SEQ + i) << 6), 0, 0);
        }
        j = i;

        // ---- A matrix (M_t), 16x4 layout: lane m holds K = kb, kb+1 ----
        // (no masking: rows 8..15 are harmless duplicates of rows 0..7)
        const f32x2 a1 = *(const f32x2*)(gp + voff);       // K = kb..kb+1
        const f32x2 a2 = *(const f32x2*)(gp + voff + 4);   // K = kb+4..kb+5

        // ---- B matrix from acc (C/D layout -> 4x16 B layout) ----
        // Low half-wave wants acc rows {0,1}/{4,5} at its OWN lane (identity,
        // no permute). High half-wave gathers rows {2,3}/{6,7} from lane-16.
        const float h0 = __shfl(acc[2], mrow, 32);
        const float h1 = __shfl(acc[3], mrow, 32);
        const float h2 = __shfl(acc[6], mrow, 32);
        const float h3 = __shfl(acc[7], mrow, 32);

        f32x2 b1, b2;
        b1.x = hi ? h0 : acc[0];   // B row kb
        b1.y = hi ? h1 : acc[1];   // B row kb+1
        b2.x = hi ? h2 : acc[4];   // B row kb+4
        b2.y = hi ? h3 : acc[5];   // B row kb+5
        // (no col masking: cols 8..15 garbage stays confined to D cols 8..15)

        // ---- hol_new = A_lo*B_lo + A_hi*B_hi  (replaces hol) ----
        f32x8 zc = {0.f, 0.f, 0.f, 0.f, 0.f, 0.f, 0.f, 0.f};
        f32x8 tmp = __builtin_amdgcn_wmma_f32_16x16x4_f32(
            false, a1, false, b1, (short)0, zc, false, false);
        acc = __builtin_amdgcn_wmma_f32_16x16x4_f32(
            false, a2, false, b2, (short)0, tmp, false, false);
    }

    // ---- store valid 8x8 block: out[b][l][r][c], c = lane ----
    if (lane < 8) {
        float* op = out + (((size_t)b * N_LOOPS + l) << 6) + lane;
#pragma unroll
        for (int r = 0; r < 8; ++r)
            op[r * KDIM] = acc[r];
    }
}

extern "C" void kernel_launch(void* const* d_in, const int* in_sizes, int n_in,
                              void* d_out, int out_size, void* d_ws, size_t ws_size,
                              hipStream_t stream) {
    const float* Gamma = (const float*)d_in[0];
    const int*   loops = (const int*)d_in[1];
    float*       out   = (float*)d_out;

    // 256 waves total = 4 batches * 64 loops; 8 waves (256 threads) per block.
    holonomy_wmma_kernel<<<dim3(BATCH * N_LOOPS / 8), dim3(256), 0, stream>>>(
        Gamma, loops, out);
}